// TimeSeriesSimilarityLoss_57999238365185
// MI455X (gfx1250) — compile-verified
//
#include <hip/hip_runtime.h>

// Single persistent workgroup (1024 threads = 32 waves) using 316KB of the
// 320KB WGP LDS. TDM DMAs `targets` into LDS while we analyze `outputs`.
#define NTHREADS 1024
#define SEG 64                 // 65536 / 1024

typedef unsigned int v4u __attribute__((ext_vector_type(4)));
typedef int          v8i __attribute__((ext_vector_type(8)));
typedef int          v4i __attribute__((ext_vector_type(4)));

// LDS layout (byte offsets into the dynamic shared blob)
#define OFF_TY   0        // float targets[65536]            262144 B
#define OFF_EPX  262144   // 2-bit packed ep for outputs      16384 B
#define OFF_EPY  278528   // 2-bit packed ep for targets      16384 B
#define OFF_FX   294912   // fluct bits, outputs               8192 B
#define OFF_FY   303104   // fluct bits, targets               8192 B
#define OFF_MY   311296   // matchedY bits                     8192 B
#define OFF_SC   319488   // int scan scratch[1024]            4096 B
#define LDS_TOTAL 323584

__device__ __forceinline__ int unpack_ep(const unsigned* ep, int i) {
  unsigned c = (ep[i >> 4] >> ((i & 15) * 2)) & 3u;
  return (c == 1u) ? 1 : ((c == 2u) ? -1 : 0);
}
__device__ __forceinline__ int getbit(const unsigned* b, int i) {
  return (int)((b[i >> 5] >> (i & 31)) & 1u);
}

// Inclusive Hillis-Steele max scan over 1024 per-segment values; returns the
// EXCLUSIVE carry (max of all earlier segments, or -1).
__device__ __forceinline__ int scan_max_excl(int v, int* sc, int tid) {
  sc[tid] = v;
  __syncthreads();
  for (int off = 1; off < NTHREADS; off <<= 1) {
    int o = (tid >= off) ? sc[tid - off] : -1;
    __syncthreads();
    if (o > sc[tid]) sc[tid] = o;
    __syncthreads();
  }
  int carry = (tid > 0) ? sc[tid - 1] : -1;
  __syncthreads();   // safe to reuse sc afterwards
  return carry;
}

// Full _analyze() for one series. USE_LDS selects the LDS-staged copy.
template <bool USE_LDS>
__device__ void analyze(const float* __restrict__ g, const float* lds_s,
                        unsigned* epArr, unsigned* fArr, int* sc,
                        int n, int tid, float thr) {
  const int base = tid * SEG;
  auto S = [&](int i) -> float { return USE_LDS ? lds_s[i] : g[i]; };

  // ---- Phase A: extreme points (3-pt stencil), per-segment last extreme ----
  unsigned epw[SEG / 16];
  for (int w = 0; w < SEG / 16; ++w) epw[w] = 0u;
  int lastExt = -1;
  float sm1 = (base > 0) ? S(base - 1) : 0.f;
  float s0  = S(base);
  for (int k = 0; k < SEG; ++k) {
    int i = base + k;
    float sp1 = (i + 1 < n) ? S(i + 1) : 0.f;
    int a;
    if (i == 0)          a = (s0 > sp1) ? 1 : -1;
    else if (i == n - 1) a = (s0 > sm1) ? 1 : -1;
    else {
      float ld = s0 - sm1, rd = s0 - sp1;
      a = (ld * rd > 0.f) ? ((ld > 0.f) ? 1 : -1) : 0;
    }
    unsigned c = (a == 1) ? 1u : ((a == -1) ? 2u : 0u);
    epw[k >> 4] |= c << ((k & 15) * 2);
    if (a != 0) lastExt = i;
    sm1 = s0; s0 = sp1;
  }
  for (int w = 0; w < SEG / 16; ++w) epArr[(base >> 4) + w] = epw[w];
  const int carryExt = scan_max_excl(lastExt, sc, tid);  // barriers inside

  // ---- Phase B: candidate flags, per-segment last candidate ----
  int runExt = carryExt, lastCand = -1;
  for (int k = 0; k < SEG; ++k) {
    int i = base + k;
    unsigned c = (epw[k >> 4] >> ((k & 15) * 2)) & 3u;
    int a = (c == 1u) ? 1 : ((c == 2u) ? -1 : 0);
    if (a != 0) {
      bool cand = (runExt < 0) || (fabsf(S(i) - S(runExt)) > thr);
      if (cand) lastCand = i;
      runExt = i;
    }
  }
  const int carryCand = scan_max_excl(lastCand, sc, tid);

  // ---- Phase C: own flag per candidate (A|B; first candidate -> 1) ----
  unsigned fb[SEG / 32] = {0u, 0u};
  runExt = carryExt;
  int runCand = carryCand;
  for (int k = 0; k < SEG; ++k) {
    int i = base + k;
    unsigned c = (epw[k >> 4] >> ((k & 15) * 2)) & 3u;
    int a = (c == 1u) ? 1 : ((c == 2u) ? -1 : 0);
    if (a != 0) {
      bool cand = (runExt < 0) || (fabsf(S(i) - S(runExt)) > thr);
      if (cand) {
        int own, p = runCand;
        if (p < 0) own = 1;                     // first candidate == index 0
        else {
          int ap = unpack_ep(epArr, p);
          bool opp = (a * ap) == -1;
          float vi = S(i), vp = S(p);
          bool stronger = (a == 1) ? (vi > vp) : (vi < vp);
          own = (opp || stronger) ? 1 : 0;
        }
        if (own) fb[k >> 5] |= 1u << (k & 31);
        runCand = i;
      }
      runExt = i;
    }
  }
  fArr[(base >> 5) + 0] = fb[0];
  fArr[(base >> 5) + 1] = fb[1];
  __syncthreads();   // all `own` bits visible before overrides

  // ---- Phase D: successor overrides prev candidate (B -> 0, C -> 1) ----
  runExt = carryExt; runCand = carryCand;
  for (int k = 0; k < SEG; ++k) {
    int i = base + k;
    unsigned c = (epw[k >> 4] >> ((k & 15) * 2)) & 3u;
    int a = (c == 1u) ? 1 : ((c == 2u) ? -1 : 0);
    if (a != 0) {
      bool cand = (runExt < 0) || (fabsf(S(i) - S(runExt)) > thr);
      if (cand) {
        int p = runCand;
        if (p >= 0) {
          int ap = unpack_ep(epArr, p);
          if ((a * ap) != -1) {                 // same orientation
            float vi = S(i), vp = S(p);
            bool stronger = (a == 1) ? (vi > vp) : (vi < vp);
            unsigned bit = 1u << (p & 31);
            if (stronger) atomicAnd(&fArr[p >> 5], ~bit);   // B: prev -> 0
            else          atomicOr (&fArr[p >> 5],  bit);   // C: prev -> 1
          }
        }
        runCand = i;
      }
      runExt = i;
    }
  }
  __syncthreads();
}

extern "C" __global__ void __launch_bounds__(NTHREADS, 1)
tsloss_kernel(const float* __restrict__ outputs, const float* __restrict__ targets,
              float* __restrict__ out, int n) {
  extern __shared__ unsigned char smem[];
  float*    tY  = (float*)(smem + OFF_TY);
  unsigned* epx = (unsigned*)(smem + OFF_EPX);
  unsigned* epy = (unsigned*)(smem + OFF_EPY);
  unsigned* fx  = (unsigned*)(smem + OFF_FX);
  unsigned* fy  = (unsigned*)(smem + OFF_FY);
  unsigned* mY  = (unsigned*)(smem + OFF_MY);
  int*      sc  = (int*)(smem + OFF_SC);
  const int tid = threadIdx.x;

  // zero matchedY bits (2 dwords per thread)
  mY[tid * 2 + 0] = 0u;
  mY[tid * 2 + 1] = 0u;

  // Kick the Tensor Data Mover: DMA targets (256 KB, 2D tile 1024x64 fp32)
  // into LDS while the ALUs analyze `outputs` from global/L2.
  if (tid == 0) {
    unsigned long long ga = (unsigned long long)(const void*)targets;
    v4u g0 = { 1u,                                        // count=1 user D#
               (unsigned)OFF_TY,                          // lds_addr
               (unsigned)(ga & 0xFFFFFFFFull),            // global_addr lo
               (unsigned)(((ga >> 32) & 0x1FFFFFFull) | 0x80000000u) }; // hi | type=2
    v8i g1 = { (int)0x20000,          // data_size = 4 B
               (int)(1024u << 16),    // tensor_dim0 = 1024 (lo16 in [63:48])
               (int)(64u   << 16),    // dim0 hi=0 | tensor_dim1 = 64
               (int)(1024u << 16),    // dim1 hi=0 | tile_dim0 = 1024
               64,                    // tile_dim1 = 64 | tile_dim2 = 0
               1024,                  // tensor_dim0_stride = 1024
               0, 0 };
    v4i z4 = { 0, 0, 0, 0 };
    v8i z8 = { 0, 0, 0, 0, 0, 0, 0, 0 };
    // clang-23 / therock-10.0 lane: 6-arg form
    // (g0: v4u, g1: v8i, g2: v4i, g3: v4i, extra: v8i, cpol: i32)
    __builtin_amdgcn_tensor_load_to_lds(g0, g1, z4, z4, z8, 0);
  }

  // Analyze outputs (X) straight from global memory — overlaps the TDM DMA.
  analyze<false>(outputs, nullptr, epx, fx, sc, n, tid, 0.5f);

  // Issuing wave drains TENSORcnt; barrier publishes the LDS copy to all waves.
  if (tid < 32) __builtin_amdgcn_s_wait_tensorcnt((short)0);
  __syncthreads();

  // Analyze targets (Y) from the LDS-staged copy.
  analyze<true>(nullptr, tY, epy, fy, sc, n, tid, 0.5f);

  // ---- Sequential greedy windowed matching on wave 0 (wave32 ballot) ----
  // Lane k owns window offset offs[k] = [0,-1,1,-2,2,-3,3,-4,4,-5,5]; lane
  // order equals the reference priority order, so ctz(ballot) == argmax(valid).
  if (tid < 32) {
    const int lane = tid;
    int off = 0;
    if (lane > 0 && lane < 11) {
      int d = (lane + 1) >> 1;
      off = (lane & 1) ? -d : d;
    }
    const bool laneActive = (lane < 11);
    float acc = 0.f;
    for (int ib = 0; ib < n; ib += 32) {
      // 32-wide skip of non-fluctuation positions
      bool f = getbit(fx, ib + lane) != 0;
      unsigned fm = __builtin_amdgcn_ballot_w32(f);
      while (fm) {
        int i = ib + (int)__builtin_ctz(fm);
        fm &= fm - 1;
        int ax = unpack_ep(epx, i);
        int j = i + off;
        bool valid = false;
        if (laneActive && j >= 0 && j < n) {
          valid = getbit(fy, j) && !getbit(mY, j) && (unpack_ep(epy, j) == ax);
        }
        unsigned vm = __builtin_amdgcn_ballot_w32(valid);
        if (vm) {
          int k = (int)__builtin_ctz(vm);       // highest-priority offset
          int dk = (k + 1) >> 1;
          int offk = (k == 0) ? 0 : ((k & 1) ? -dk : dk);
          int js = i + offk;
          if (lane == 0) {
            mY[js >> 5] |= 1u << (js & 31);     // mark matched
            float d2 = outputs[i] - tY[js];
            acc += d2 * d2;
          }
        }
      }
    }
    if (lane == 0) out[0] = acc;
  }
}

extern "C" void kernel_launch(void* const* d_in, const int* in_sizes, int n_in,
                              void* d_out, int out_size, void* d_ws, size_t ws_size,
                              hipStream_t stream) {
  (void)n_in; (void)out_size; (void)d_ws; (void)ws_size;
  const float* outputs = (const float*)d_in[0];
  const float* targets = (const float*)d_in[1];
  float* out = (float*)d_out;
  int n = in_sizes[0];   // 65536

  (void)hipFuncSetAttribute((const void*)tsloss_kernel,
                            hipFuncAttributeMaxDynamicSharedMemorySize, LDS_TOTAL);
  tsloss_kernel<<<1, NTHREADS, LDS_TOTAL, stream>>>(outputs, targets, out, n);
}